// SimpleMoERouter_54219667144993
// MI455X (gfx1250) — compile-verified
//
#include <hip/hip_runtime.h>

typedef __attribute__((ext_vector_type(2))) float v2f;
typedef __attribute__((ext_vector_type(8))) float v8f;

#define D_MODEL     4096
#define NUM_EXPERTS 64
#define MB          64           // tokens per block
#define KC          64           // K-chunk staged in LDS
#define LDA         (KC + 4)     // padded row stride: bank = (4*row + k) % 64, conflict-free
#define LDB         (KC + 4)
#define LDL         (KC + 4)     // logits stride (overlaid on sA, 64 x 68 fits exactly)

// 128 threads = 4 waves. Block handles 64 tokens x 64 experts.
// Wave w owns M-tile w (16 tokens) and keeps 4 N-tile accumulators in registers,
// so each A fragment load is amortized over 4 WMMAs.
__global__ __launch_bounds__(128) void moe_router_kernel(
    const float* __restrict__ hs,   // [tokens, 4096]
    const float* __restrict__ W,    // [64, 4096]
    float* __restrict__ out_w,      // [tokens, 2]
    int*   __restrict__ out_i)      // [tokens, 2]
{
    __shared__ float sA[MB * LDA];           // 17408 B
    __shared__ float sB[NUM_EXPERTS * LDB];  // 17408 B

    const int tid  = threadIdx.x;
    const int wave = tid >> 5;
    const int lane = tid & 31;
    const int m0   = blockIdx.x * MB;

    v8f acc0 = {}, acc1 = {}, acc2 = {}, acc3 = {};

    const int row   = lane & 15;
    const int khalf = (lane >> 4) << 1;      // 0 or 2 (upper half-wave -> K+2)

    for (int k0 = 0; k0 < D_MODEL; k0 += KC) {
        __syncthreads();
        // ---- stage A: 64 rows x 64 floats = 1024 float4, 8 per thread ----
        #pragma unroll
        for (int i = 0; i < 8; ++i) {
            int idx = tid + i * 128;         // 0..1023
            int r   = idx >> 4;              // 16 float4 per row
            int c   = (idx & 15) << 2;
            float4 v = *(const float4*)&hs[(size_t)(m0 + r) * D_MODEL + k0 + c];
            *(float4*)&sA[r * LDA + c] = v;
        }
        // ---- stage B: 64 experts x 64 floats (L2-resident W) ----
        #pragma unroll
        for (int i = 0; i < 8; ++i) {
            int idx = tid + i * 128;
            int r   = idx >> 4;
            int c   = (idx & 15) << 2;
            float4 v = *(const float4*)&W[(size_t)r * D_MODEL + k0 + c];
            *(float4*)&sB[r * LDB + c] = v;
        }
        // prefetch next A chunk (global_prefetch_b8)
        if (tid < MB && (k0 + KC) < D_MODEL) {
            __builtin_prefetch(&hs[(size_t)(m0 + tid) * D_MODEL + k0 + KC], 0, 1);
        }
        __syncthreads();

        // ---- WMMA loop: 16 k-steps x 4 N-tiles = 64 wmma per wave per chunk ----
        const float* aRow = &sA[(wave * 16 + row) * LDA];
        const float* bRow = &sB[row * LDB];
        #pragma unroll
        for (int kk = 0; kk < KC; kk += 4) {
            const int ko = kk + khalf;
            v2f a  = *(const v2f*)&aRow[ko];                 // A(m, ko..ko+1), reused 4x
            v2f b0 = *(const v2f*)&bRow[ko];                 // experts  0..15
            v2f b1 = *(const v2f*)&bRow[16 * LDB + ko];      // experts 16..31
            v2f b2 = *(const v2f*)&bRow[32 * LDB + ko];      // experts 32..47
            v2f b3 = *(const v2f*)&bRow[48 * LDB + ko];      // experts 48..63
            acc0 = __builtin_amdgcn_wmma_f32_16x16x4_f32(false, a, false, b0, (short)0, acc0, false, false);
            acc1 = __builtin_amdgcn_wmma_f32_16x16x4_f32(false, a, false, b1, (short)0, acc1, false, false);
            acc2 = __builtin_amdgcn_wmma_f32_16x16x4_f32(false, a, false, b2, (short)0, acc2, false, false);
            acc3 = __builtin_amdgcn_wmma_f32_16x16x4_f32(false, a, false, b3, (short)0, acc3, false, false);
        }
    }

    // ---- scatter logits: overlay sL on sA (dead after last compute barrier) ----
    __syncthreads();
    float* sL = sA;                          // 64 tokens x LDL, 64*68 == MB*LDA
    {
        const int mh = (lane >> 4) << 3;     // C layout: VGPR r <-> M = r + 8*(lane>=16)
        const int n  = lane & 15;
        const int mb = wave * 16 + mh;
        #pragma unroll
        for (int r = 0; r < 8; ++r) {
            sL[(mb + r) * LDL + (n +  0)] = acc0[r];
            sL[(mb + r) * LDL + (n + 16)] = acc1[r];
            sL[(mb + r) * LDL + (n + 32)] = acc2[r];
            sL[(mb + r) * LDL + (n + 48)] = acc3[r];
        }
    }
    __syncthreads();

    // ---- top-2 + softmax, one thread per token (64 tokens) ----
    if (tid < MB) {
        const float* lg = &sL[tid * LDL];
        float best0 = lg[0]; int idx0 = 0;
        for (int e = 1; e < NUM_EXPERTS; ++e) {   // strict > keeps lowest index on tie
            float v = lg[e];
            if (v > best0) { best0 = v; idx0 = e; }
        }
        float best1 = -3.402823466e+38f; int idx1 = 0;
        for (int e = 0; e < NUM_EXPERTS; ++e) {
            if (e == idx0) continue;
            float v = lg[e];
            if (v > best1) { best1 = v; idx1 = e; }
        }
        float e1  = __expf(best1 - best0);        // best0 >= best1
        float inv = 1.0f / (1.0f + e1);
        int t = m0 + tid;
        out_w[t * 2 + 0] = inv;
        out_w[t * 2 + 1] = e1 * inv;
        out_i[t * 2 + 0] = idx0;
        out_i[t * 2 + 1] = idx1;
    }
}

extern "C" void kernel_launch(void* const* d_in, const int* in_sizes, int n_in,
                              void* d_out, int out_size, void* d_ws, size_t ws_size,
                              hipStream_t stream) {
    const float* hs = (const float*)d_in[0];   // [B*S, 4096] fp32
    const float* W  = (const float*)d_in[1];   // [64, 4096]  fp32
    const int tokens = in_sizes[0] / D_MODEL;  // 16384

    float* out_w = (float*)d_out;                        // [tokens,2] fp32
    int*   out_i = (int*)d_out + (size_t)tokens * 2;     // [tokens,2] int32

    const int blocks = tokens / MB;            // 256
    hipLaunchKernelGGL(moe_router_kernel, dim3(blocks), dim3(128), 0, stream,
                       hs, W, out_w, out_i);
}